// FPSModule_49495203119342
// MI455X (gfx1250) — compile-verified
//
#include <hip/hip_runtime.h>
#include <hip/hip_bf16.h>
#include <stdint.h>
#include <stddef.h>

// Problem constants (match reference)
#define B_       8
#define N_       65536
#define C_       256
#define NPOINT_  1024

// FPS kernel geometry: 1 workgroup per batch, 32 waves (wave32) = 1024 threads
#define THREADS_      1024
#define WAVES_        32
#define PTS_PER_WAVE  (N_ / WAVES_)   // 2048 points owned by each wave
#define CHUNK_PTS     256             // points staged per chunk per wave
#define CHUNK_BYTES   (CHUNK_PTS * 12)// 3072 B, 16B-aligned, = 192 b128 units
#define NCHUNK        (PTS_PER_WAVE / CHUNK_PTS)  // 8
#define PTS_PER_LANE_PER_CHUNK (CHUNK_PTS / 32)   // 8
#define ASYNC_OPS_PER_CHUNK (CHUNK_BYTES / 16 / 32) // 6 b128 issues per lane

// ---- CDNA5 async global->LDS path (guarded; fallback = direct L2 loads) ----
#if defined(__has_builtin)
#  if __has_builtin(__builtin_amdgcn_global_load_async_to_lds_b128) && \
      __has_builtin(__builtin_amdgcn_s_wait_asynccnt)
#    define USE_ASYNC 1
#  endif
#endif
#ifndef USE_ASYNC
#  define USE_ASYNC 0
#endif

#if USE_ASYNC
// Exact parameter types per hipcc diagnostic:
//   param 1: 'int __attribute__((vector_size(16))) __device__ *'  (AS1, global)
//   param 2: LDS-side analogue (AS3, shared)
typedef int v4i_ __attribute__((vector_size(4 * sizeof(int))));
typedef __attribute__((address_space(1))) v4i_* gv4_t;
typedef __attribute__((address_space(3))) v4i_* lv4_t;

__device__ __forceinline__ void async_copy16(const void* g, void* l) {
  // per-lane: copies 16 bytes from global address g to LDS address l, tracked on ASYNCcnt
  __builtin_amdgcn_global_load_async_to_lds_b128((gv4_t)g, (lv4_t)l, 0, 0);
}
#endif

// ---------------------------------------------------------------------------
// FPS: one workgroup per batch. Running min-distances live in VGPRs (64/thread).
// xyz is streamed per round through wave-private LDS double buffers using
// async global->LDS copies; only the argmax reduction needs barriers.
// ---------------------------------------------------------------------------
__global__ __launch_bounds__(THREADS_) void fps_kernel(
    const float* __restrict__ xyz,      // (B, N, 3)
    float* __restrict__ out_xyz,        // (B, NPOINT, 3)
    int*   __restrict__ out_inds)       // (B, NPOINT)
{
  const int b    = blockIdx.x;
  const int tid  = threadIdx.x;
  const int lane = tid & 31;
  const int wv   = tid >> 5;
  const float* __restrict__ xb = xyz + (size_t)b * N_ * 3;

#if USE_ASYNC
  __shared__ float stage[WAVES_ * 2 * (CHUNK_BYTES / 4)];  // 192 KB, wave-private slices
#endif
  __shared__ float bcast[3];
  __shared__ int   bidx;
  __shared__ float pval[WAVES_];
  __shared__ int   pidx[WAVES_];

  // running min squared distance for the 64 points this thread owns (in VGPRs)
  float dist[NCHUNK * PTS_PER_LANE_PER_CHUNK];
#pragma unroll
  for (int i = 0; i < NCHUNK * PTS_PER_LANE_PER_CHUNK; ++i) dist[i] = 1e10f;

  if (tid == 0) {  // deterministic analogue of the random initial index
    bcast[0] = xb[0]; bcast[1] = xb[1]; bcast[2] = xb[2]; bidx = 0;
  }
  __syncthreads();

  const int wbase = wv * PTS_PER_WAVE;

  for (int it = 0; it < NPOINT_; ++it) {
    // record the current farthest point (selected at end of previous round)
    if (tid == 0) {
      out_inds[b * NPOINT_ + it] = bidx;
      float* o = out_xyz + ((size_t)b * NPOINT_ + it) * 3;
      o[0] = bcast[0]; o[1] = bcast[1]; o[2] = bcast[2];
    }
    const float cx = bcast[0], cy = bcast[1], cz = bcast[2];

    float bestV = -1.0f;
    int   bestI = 0;

#if USE_ASYNC
    // prologue: stage chunks 0 and 1 into this wave's two LDS buffers
    {
      const char* g0 = (const char*)(xb + (size_t)wbase * 3);
#pragma unroll
      for (int cc = 0; cc < 2; ++cc) {
        const char* gsrc = g0 + cc * CHUNK_BYTES;
        char* ldst = (char*)&stage[(wv * 2 + cc) * (CHUNK_BYTES / 4)];
#pragma unroll
        for (int j = 0; j < ASYNC_OPS_PER_CHUNK; ++j) {
          const int off = (j * 32 + lane) * 16;
          async_copy16(gsrc + off, ldst + off);
        }
      }
    }
#endif

#pragma unroll
    for (int c = 0; c < NCHUNK; ++c) {
#if USE_ASYNC
      // async loads complete in order: waiting to <=6 retires chunk c's copies.
      if (c == NCHUNK - 1) __builtin_amdgcn_s_wait_asynccnt(0);
      else                 __builtin_amdgcn_s_wait_asynccnt(ASYNC_OPS_PER_CHUNK);
      const float* buf = &stage[(wv * 2 + (c & 1)) * (CHUNK_BYTES / 4)];
#endif
#pragma unroll
      for (int m = 0; m < PTS_PER_LANE_PER_CHUNK; ++m) {
        const int pl = lane + 32 * m;                    // point within chunk
        const int gi = wbase + c * CHUNK_PTS + pl;       // global point index
        float px, py, pz;
#if USE_ASYNC
        px = buf[pl * 3 + 0]; py = buf[pl * 3 + 1]; pz = buf[pl * 3 + 2];
#else
        const float* gp = xb + (size_t)gi * 3;
        px = gp[0]; py = gp[1]; pz = gp[2];
#endif
        const float dx = px - cx, dy = py - cy, dz = pz - cz;
        const float dd = dx * dx + dy * dy + dz * dz;
        float dc = dist[c * PTS_PER_LANE_PER_CHUNK + m];
        dc = fminf(dc, dd);
        dist[c * PTS_PER_LANE_PER_CHUNK + m] = dc;
        if (dc > bestV) { bestV = dc; bestI = gi; }
      }
#if USE_ASYNC
      if (c + 2 < NCHUNK) {
        // ensure our ds reads of this buffer retired before overwriting it
        asm volatile("s_wait_dscnt 0" ::: "memory");
        const char* gsrc = (const char*)(xb + (size_t)(wbase + (c + 2) * CHUNK_PTS) * 3);
        char* ldst = (char*)&stage[(wv * 2 + (c & 1)) * (CHUNK_BYTES / 4)];
#pragma unroll
        for (int j = 0; j < ASYNC_OPS_PER_CHUNK; ++j) {
          const int off = (j * 32 + lane) * 16;
          async_copy16(gsrc + off, ldst + off);
        }
      }
#endif
    }

    // wave32 argmax reduction (ties -> lower index, matching argmax-first)
#pragma unroll
    for (int off = 16; off > 0; off >>= 1) {
      const float ov = __shfl_xor(bestV, off, 32);
      const int   oi = __shfl_xor(bestI, off, 32);
      if (ov > bestV || (ov == bestV && oi < bestI)) { bestV = ov; bestI = oi; }
    }
    if (lane == 0) { pval[wv] = bestV; pidx[wv] = bestI; }
    __syncthreads();
    if (wv == 0) {
      float v = pval[lane];
      int   i2 = pidx[lane];
#pragma unroll
      for (int off = 16; off > 0; off >>= 1) {
        const float ov = __shfl_xor(v, off, 32);
        const int   oi = __shfl_xor(i2, off, 32);
        if (ov > v || (ov == v && oi < i2)) { v = ov; i2 = oi; }
      }
      if (lane == 0) {
        bidx = i2;
        const float* p = xb + (size_t)i2 * 3;
        bcast[0] = p[0]; bcast[1] = p[1]; bcast[2] = p[2];
      }
    }
    __syncthreads();
  }
}

// ---------------------------------------------------------------------------
// Feature gather: one block per (b, c) row; scattered reads (inherent),
// fully coalesced writes.
// ---------------------------------------------------------------------------
__global__ __launch_bounds__(256) void gather_features_kernel(
    const float* __restrict__ feat,     // (B, C, N)
    const int*   __restrict__ inds,     // (B, NPOINT)
    float*       __restrict__ out_feat) // (B, C, NPOINT)
{
  const int bc = blockIdx.x;            // 0 .. B*C-1
  const int bq = bc / C_;
  const float* __restrict__ f = feat + (size_t)bc * N_;
  float* __restrict__ o = out_feat + (size_t)bc * NPOINT_;
  const int* __restrict__ ib = inds + bq * NPOINT_;
  for (int j = threadIdx.x; j < NPOINT_; j += 256) {
    o[j] = f[ib[j]];
  }
}

extern "C" void kernel_launch(void* const* d_in, const int* in_sizes, int n_in,
                              void* d_out, int out_size, void* d_ws, size_t ws_size,
                              hipStream_t stream) {
  (void)in_sizes; (void)n_in; (void)d_ws; (void)ws_size; (void)out_size;
  const float* xyz  = (const float*)d_in[0];   // (B, N, 3) fp32
  const float* feat = (const float*)d_in[1];   // (B, C, N) fp32
  // d_in[2] = npoint scalar; fixed at compile time.

  float* out      = (float*)d_out;
  float* out_xyz  = out;                                        // B*NPOINT*3
  float* out_feat = out + (size_t)B_ * NPOINT_ * 3;             // B*C*NPOINT
  int*   out_inds = (int*)(out + (size_t)B_ * NPOINT_ * 3
                               + (size_t)B_ * C_ * NPOINT_);    // B*NPOINT (int32 bits)

  hipLaunchKernelGGL(fps_kernel, dim3(B_), dim3(THREADS_), 0, stream,
                     xyz, out_xyz, out_inds);
  hipLaunchKernelGGL(gather_features_kernel, dim3(B_ * C_), dim3(256), 0, stream,
                     feat, out_inds, out_feat);
}